// FeatureExtractor_87247965651637
// MI455X (gfx1250) — compile-verified
//
#include <hip/hip_runtime.h>
#include <math.h>

typedef __attribute__((ext_vector_type(2))) float v2f;
typedef __attribute__((ext_vector_type(8))) float v8f;

// ---------------------------------------------------------------------------
// Kernel 0: compute Butterworth bandpass coefficients (order 5 -> 11 taps)
// exactly mirroring the numpy zpk pipeline, in double, cast to float.
// coefs layout: band g: [0..10]=b, [11..21]=a   (6*22 floats)
// ---------------------------------------------------------------------------
__global__ void butter_coeff_kernel(float* __restrict__ coefs) {
    int g = threadIdx.x;
    if (g >= 6) return;
    const double los[6] = {8.0, 12.0, 16.0, 20.0, 24.0, 28.0};
    const double his[6] = {12.0, 16.0, 20.0, 24.0, 28.0, 32.0};
    const double PI = 3.14159265358979323846;
    double lo = los[g], hi = his[g];
    // bilinear prewarp (fs2 = 2): warped = 4*tan(pi*Wn/2), Wn = f/125
    double w0 = 4.0 * tan(PI * (lo / 125.0) / 2.0);
    double w1 = 4.0 * tan(PI * (hi / 125.0) / 2.0);
    double bw = w1 - w0;
    double wo = sqrt(w0 * w1);
    // analog prototype poles: p = -exp(i*pi*m/10), m = -4,-2,0,2,4
    double pr[5], pim[5];
    for (int i = 0; i < 5; i++) {
        double th = PI * (double)(-4 + 2 * i) / 10.0;
        pr[i] = -cos(th);
        pim[i] = -sin(th);
    }
    // lp2bp: p_lp = p*bw/2 ; p_bp = p_lp +/- sqrt(p_lp^2 - wo^2)
    double br[10], bi[10];
    for (int i = 0; i < 5; i++) {
        double ar = pr[i] * bw * 0.5, ai = pim[i] * bw * 0.5;
        double sr = ar * ar - ai * ai - wo * wo;
        double si = 2.0 * ar * ai;
        double r = sqrt(sr * sr + si * si);
        double qr = sqrt(0.5 * (r + sr));
        double qi = sqrt(0.5 * (r - sr));
        if (si < 0.0) qi = -qi;                 // principal sqrt branch
        br[i] = ar + qr;      bi[i] = ai + qi;
        br[i + 5] = ar - qr;  bi[i + 5] = ai - qi;
    }
    // bilinear (fs2x = 4): p_d = (4+p)/(4-p); P = prod(4 - p_bp)
    double dr[10], di[10], Pr = 1.0, Pi = 0.0;
    for (int i = 0; i < 10; i++) {
        double nr = 4.0 + br[i], ni = bi[i];
        double mr = 4.0 - br[i], mi = -bi[i];
        double den = mr * mr + mi * mi;
        dr[i] = (nr * mr + ni * mi) / den;
        di[i] = (ni * mr - nr * mi) / den;
        double tr = Pr * mr - Pi * mi;
        double ti = Pr * mi + Pi * mr;
        Pr = tr; Pi = ti;
    }
    double k_bp = bw * bw * bw * bw * bw;
    // k_d = k_bp * Re(4^5 / P)
    double k_d = k_bp * (1024.0 * Pr / (Pr * Pr + Pi * Pi));
    // a = Re(poly(p_d))
    double cr[11], ci[11];
    cr[0] = 1.0; ci[0] = 0.0;
    for (int i = 1; i < 11; i++) { cr[i] = 0.0; ci[i] = 0.0; }
    for (int i = 0; i < 10; i++) {
        for (int k = i + 1; k >= 1; k--) {
            double ncr = cr[k] - (dr[i] * cr[k - 1] - di[i] * ci[k - 1]);
            double nci = ci[k] - (dr[i] * ci[k - 1] + di[i] * cr[k - 1]);
            cr[k] = ncr; ci[k] = nci;
        }
    }
    // b = k_d * poly([1 x5, -1 x5]) = k_d * coeffs of (z^2-1)^5
    const double pat[11] = {1, 0, -5, 0, 10, 0, -10, 0, 5, 0, -1};
    for (int n = 0; n < 11; n++) {
        coefs[g * 22 + n] = (float)(k_d * pat[n]);
        coefs[g * 22 + 11 + n] = (float)cr[n];
    }
}

// ---------------------------------------------------------------------------
// Kernel 1: per-batch CBAM + channel-mix (filter/mix commute!) + IIR + var.
// One block per batch element. Writes V[b, 1200] (12 signals x 100 windows).
// ---------------------------------------------------------------------------
#define C_CH 22
#define TLEN 1000
#define MIXP 1001   // odd stride -> no LDS bank conflicts across the 12 rows

__launch_bounds__(256)
__global__ void feat_kernel(const float* __restrict__ x,
                            const float* __restrict__ w1,
                            const float* __restrict__ w2,
                            const float* __restrict__ convw,
                            const float* __restrict__ dw,
                            const float* __restrict__ coefs,
                            float* __restrict__ V) {
    __shared__ float s_mix[12 * MIXP];   // mixed signals s[j][t]
    __shared__ float s_sfac[TLEN];       // sigmoid(spatial attention)
    __shared__ float s_avg[TLEN];
    __shared__ float s_max[TLEN];
    __shared__ float s_csum[C_CH], s_cmax[C_CH];
    __shared__ float s_h[11];
    __shared__ float s_ca[C_CH];
    __shared__ float s_wm[12][C_CH];     // ca-folded mixing weights
    __shared__ float s_cf[132];          // filter coeffs

    const int b = blockIdx.x;
    const int tid = threadIdx.x;
    const int lane = tid & 31;
    const int wv = tid >> 5;
    const float* xb = x + (size_t)b * C_CH * TLEN;

    if (tid < 132) s_cf[tid] = coefs[tid];

    // --- Phase A: per-channel sum & max over T (one wave per channel set) ---
    for (int c = wv; c < C_CH; c += 8) {
        float sm = 0.f, mx = -INFINITY;
        for (int t = lane; t < TLEN; t += 32) {
            float v = xb[c * TLEN + t];
            sm += v;
            mx = fmaxf(mx, v);
        }
        for (int off = 16; off > 0; off >>= 1) {
            sm += __shfl_xor(sm, off);
            mx = fmaxf(mx, __shfl_xor(mx, off));
        }
        if (lane == 0) { s_csum[c] = sm; s_cmax[c] = mx; }
    }
    __syncthreads();

    // --- Phase B: channel-attention MLP (22 -> 11 -> 22), sigmoid ---
    if (tid < 11) {
        float ha = 0.f, hm = 0.f;
        for (int c = 0; c < C_CH; c++) {
            float wv1 = w1[c * 11 + tid];
            ha = fmaf(s_csum[c] / 1000.f, wv1, ha);
            hm = fmaf(s_cmax[c], wv1, hm);
        }
        s_h[tid] = fmaxf(ha, 0.f) + fmaxf(hm, 0.f);  // output layer is linear
    }
    __syncthreads();
    if (tid < C_CH) {
        float o = 0.f;
        for (int k = 0; k < 11; k++) o = fmaf(s_h[k], w2[k * C_CH + tid], o);
        s_ca[tid] = 1.f / (1.f + expf(-o));
    }
    __syncthreads();

    // --- Phase C: spatial stats per t, 7-tap conv (center row only), sigmoid
    for (int t = tid; t < TLEN; t += 256) {
        float mn = 0.f, mx = -INFINITY;
        for (int c = 0; c < C_CH; c++) {
            float v = xb[c * TLEN + t] * s_ca[c];
            mn += v;
            mx = fmaxf(mx, v);
        }
        s_avg[t] = mn / 22.f;
        s_max[t] = mx;
    }
    __syncthreads();
    {
        float cwm[7], cwx[7];
        for (int k = 0; k < 7; k++) {
            cwm[k] = convw[21 + k];   // conv_w[0,0,3,k] (mean channel)
            cwx[k] = convw[70 + k];   // conv_w[0,1,3,k] (max channel)
        }
        for (int t = tid; t < TLEN; t += 256) {
            float s = 0.f;
            for (int k = 0; k < 7; k++) {
                int tt = t + k - 3;
                if (tt >= 0 && tt < TLEN)
                    s += cwm[k] * s_avg[tt] + cwx[k] * s_max[tt];
            }
            s_sfac[t] = 1.f / (1.f + expf(-s));
        }
    }
    // --- Phase D: fold channel attention into mix weights ---
    for (int i = tid; i < 12 * C_CH; i += 256) {
        int j = i / C_CH, c = i % C_CH;
        s_wm[j][c] = dw[i] * s_ca[c];
    }
    __syncthreads();

    // --- Phase E: channel mixing (filter commutes with channel mix) ---
    for (int t = tid; t < TLEN; t += 256) {
        float acc[12];
        for (int j = 0; j < 12; j++) acc[j] = 0.f;
        for (int c = 0; c < C_CH; c++) {
            float v = xb[c * TLEN + t];
            for (int j = 0; j < 12; j++) acc[j] = fmaf(s_wm[j][c], v, acc[j]);
        }
        float sf = s_sfac[t];
        for (int j = 0; j < 12; j++) s_mix[j * MIXP + t] = acc[j] * sf;
    }
    __syncthreads();

    // --- Phase F: 12 IIR recurrences (DF2T) + windowed variance ---
    if (tid < 12) {
        const float* cb = &s_cf[(tid >> 1) * 22];
        const float b0 = cb[0];
        float bt[10], at[10], z[10];
        for (int i = 0; i < 10; i++) {
            bt[i] = cb[1 + i];
            at[i] = cb[12 + i];
            z[i] = 0.f;
        }
        const float* srow = &s_mix[tid * MIXP];
        float* vout = V + (size_t)b * 1200 + tid * 100;
        float ybuf[10];
        for (int w = 0; w < 100; w++) {
            float ysum = 0.f;
            for (int q = 0; q < 10; q++) {
                float xv = srow[w * 10 + q];
                float y = fmaf(b0, xv, z[0]);
                for (int i = 0; i < 9; i++)
                    z[i] = fmaf(-at[i], y, fmaf(bt[i], xv, z[i + 1]));
                z[9] = fmaf(-at[9], y, bt[9] * xv);
                ybuf[q] = y;
                ysum += y;
            }
            float mean = ysum * 0.1f;
            float var = 0.f;
            for (int q = 0; q < 10; q++) {
                float d = ybuf[q] - mean;
                var = fmaf(d, d, var);
            }
            vout[w] = var * 0.1f;   // biased variance
        }
    }
}

// ---------------------------------------------------------------------------
// Kernel 2: out[512,64] = V[512,1200] @ W[1200,64] + bias via f32 WMMA
// V_WMMA_F32_16X16X4_F32: A 16x4 (2 f32/lane), B 4x16 (2 f32/lane), C/D 8 f32.
// One wave per 16x16 tile; 32 M-tiles x 4 N-tiles = 128 waves = 32 blocks.
// ---------------------------------------------------------------------------
__launch_bounds__(128)
__global__ void fc_wmma_kernel(const float* __restrict__ V,
                               const float* __restrict__ W,
                               const float* __restrict__ bias,
                               float* __restrict__ out) {
    const int lane = threadIdx.x & 31;
    const int wv = threadIdx.x >> 5;
    const int tile = blockIdx.x * 4 + wv;
    const int mt = tile >> 2;      // 0..31
    const int nt = tile & 3;       // 0..3
    const int rowA = mt * 16 + (lane & 15);
    const int colB = nt * 16 + (lane & 15);
    const int kh = (lane < 16) ? 0 : 2;  // half-wave K split per ISA layout

    v8f acc = {};
    const float* arow = V + (size_t)rowA * 1200 + kh;
    for (int k = 0; k < 1200; k += 4) {
        v2f a = *(const v2f*)(arow + k);            // A[row, k+kh .. k+kh+1]
        v2f bv;
        bv.x = W[(size_t)(k + kh) * 64 + colB];     // B[k+kh,   col]
        bv.y = W[(size_t)(k + kh + 1) * 64 + colB]; // B[k+kh+1, col]
        acc = __builtin_amdgcn_wmma_f32_16x16x4_f32(
            false, a, false, bv, (short)0, acc, false, false);
    }
    float bb = bias[colB];
    const int mbase = mt * 16 + ((lane < 16) ? 0 : 8);
    for (int r = 0; r < 8; r++)
        out[(size_t)(mbase + r) * 64 + colB] = acc[r] + bb;
}

// ---------------------------------------------------------------------------
extern "C" void kernel_launch(void* const* d_in, const int* in_sizes, int n_in,
                              void* d_out, int out_size, void* d_ws, size_t ws_size,
                              hipStream_t stream) {
    (void)in_sizes; (void)n_in; (void)out_size; (void)ws_size;
    const float* x     = (const float*)d_in[0];  // [512,1,22,1000]
    const float* w1    = (const float*)d_in[1];  // [22,11]
    const float* w2    = (const float*)d_in[2];  // [11,22]
    const float* convw = (const float*)d_in[3];  // [1,2,7,7]
    const float* dw    = (const float*)d_in[4];  // [12,1,22,1]
    const float* fcw   = (const float*)d_in[5];  // [1200,64]
    const float* fcb   = (const float*)d_in[6];  // [64]
    float* out = (float*)d_out;                  // [512,64]

    float* coefs = (float*)d_ws;                 // 132 floats
    float* V = coefs + 256;                      // [512,1200] fp32

    butter_coeff_kernel<<<1, 32, 0, stream>>>(coefs);
    feat_kernel<<<512, 256, 0, stream>>>(x, w1, w2, convw, dw, coefs, V);
    fc_wmma_kernel<<<32, 128, 0, stream>>>(V, fcw, fcb, out);
}